// RnCLoss_41790031790868
// MI455X (gfx1250) — compile-verified
//
#include <hip/hip_runtime.h>
#include <hip/hip_bf16.h>
#include <math.h>

// CDNA5 / gfx1250: wave32, WMMA 16x16x4 f32.

typedef __attribute__((ext_vector_type(2))) float v2f;
typedef __attribute__((ext_vector_type(8))) float v8f;

#define TEMP_INV 10.0f
#define MAX_N 256
#define MAX_D 512
#define NTHREADS 256

#if defined(__has_builtin)
#if __has_builtin(__builtin_amdgcn_wmma_f32_16x16x4_f32)
#define USE_WMMA_F32 1
#endif
#endif

__device__ __forceinline__ float blockReduceSum(float v, float* s_red) {
    const int t = threadIdx.x;
    s_red[t] = v;
    __syncthreads();
    for (int off = NTHREADS / 2; off > 0; off >>= 1) {
        if (t < off) s_red[t] += s_red[t + off];
        __syncthreads();
    }
    float r = s_red[0];
    __syncthreads();
    return r;
}

__global__ __launch_bounds__(NTHREADS)
void rnc_per_sample_kernel(const float* __restrict__ feat,   // (bs,N,D)
                           const float* __restrict__ text,   // (bs,D)
                           const int*   __restrict__ nframes,// (bs,)
                           const float* __restrict__ labels, // (bs,N,1)
                           float* __restrict__ ws,           // out: loss[bs], bb[bs]
                           int N, int D) {
    __shared__ float s_red[NTHREADS];
    __shared__ float s_ip[MAX_N];
    __shared__ float s_nrm2[MAX_N];
    __shared__ float s_sv[MAX_N];
    __shared__ float s_sumf[MAX_D];

    if (N > MAX_N || D > MAX_D || (N & 15) || (D & 3)) return; // uniform

    const int b    = blockIdx.x;
    const int tid  = threadIdx.x;
    const int wave = tid >> 5;
    const int lane = tid & 31;
    const int half = lane >> 4;     // 0: lanes 0-15, 1: lanes 16-31
    const int mrow = lane & 15;     // row within 16-row tile
    const int kq   = half * 2;      // K offset within 4-wide chunk

    const size_t fbase = (size_t)b * N * D;
    const float* fb = feat + fbase;
    const float* tx = text + (size_t)b * D;
    const int n = nframes[b];

    // ---- ||text_b||^2 ----
    float tp = 0.f;
    for (int c = tid; c < D; c += NTHREADS) { float v = tx[c]; tp = fmaf(v, v, tp); }
    const float tnorm2 = blockReduceSum(tp, s_red);

    // ---- per-row: ip[i] = feat_i . text_b  (WMMA 16x16x4 f32),
    //               nrm2[i] = ||feat_i||^2   (fused per-lane partials) ----
    const int ntiles = N / 16;
    const int nwaves = NTHREADS / 32;
    for (int tile = wave; tile < ntiles; tile += nwaves) {
        const int row0 = tile * 16;
        const float* rowp = fb + (size_t)(row0 + mrow) * D;
        float np = 0.f;
#ifdef USE_WMMA_F32
        v8f c = {0.f, 0.f, 0.f, 0.f, 0.f, 0.f, 0.f, 0.f};
#pragma unroll 4
        for (int k = 0; k < D; k += 4) {
            float a0 = rowp[k + kq], a1 = rowp[k + kq + 1];
            float b0 = tx[k + kq],   b1 = tx[k + kq + 1];   // B column-broadcast
            v2f A = {a0, a1};
            v2f B = {b0, b1};
            c = __builtin_amdgcn_wmma_f32_16x16x4_f32(
                    false, A, false, B, (short)0, c, false, false);
            np = fmaf(a0, a0, fmaf(a1, a1, np));
        }
        float nfull = np + __shfl_xor(np, 16, 32);
        if (half == 0) s_nrm2[row0 + mrow] = nfull;
        // D layout: VGPR r -> M=r (lanes 0-15), M=r+8 (lanes 16-31); all N cols equal
        if (lane == 0)  { for (int r = 0; r < 8; ++r) s_ip[row0 + r]     = c[r]; }
        if (lane == 16) { for (int r = 0; r < 8; ++r) s_ip[row0 + 8 + r] = c[r]; }
#else
        float ipp = 0.f;
#pragma unroll 4
        for (int k = 0; k < D; k += 4) {
            float a0 = rowp[k + kq], a1 = rowp[k + kq + 1];
            ipp = fmaf(a0, tx[k + kq], fmaf(a1, tx[k + kq + 1], ipp));
            np  = fmaf(a0, a0, fmaf(a1, a1, np));
        }
        float nfull  = np  + __shfl_xor(np, 16, 32);
        float ipfull = ipp + __shfl_xor(ipp, 16, 32);
        if (half == 0) { s_nrm2[row0 + mrow] = nfull; s_ip[row0 + mrow] = ipfull; }
#endif
    }
    __syncthreads();

    // ---- sum_f over interior rows (coalesced: fixed i, threads sweep columns) ----
    for (int c = tid; c < D; c += NTHREADS) {
        float s = 0.f;
        for (int i = 1; i < n - 1; ++i) s += fb[(size_t)i * D + c];
        s_sumf[c] = s;
    }
    // S2 = sum over interior of ||feat_i||^2
    float s2p = (tid >= 1 && tid < n - 1) ? s_nrm2[tid] : 0.f;
    const float S2 = blockReduceSum(s2p, s_red);

    // ---- scalar dots: P0 = f0.sum_f, Pe = fe.sum_f, n0e = f0.fe ----
    const float* f0 = fb;
    const float* fe = fb + (size_t)(n - 1) * D;
    float p0p = 0.f, pep = 0.f, noep = 0.f;
    for (int c = tid; c < D; c += NTHREADS) {
        float v0 = f0[c], ve = fe[c], sf = s_sumf[c];
        p0p  = fmaf(v0, sf, p0p);
        pep  = fmaf(ve, sf, pep);
        noep = fmaf(v0, ve, noep);
    }
    const float P0  = blockReduceSum(p0p, s_red);
    const float Pe  = blockReduceSum(pep, s_red);
    const float N0E = blockReduceSum(noep, s_red);
    const float n00 = s_nrm2[0];
    const float nee = s_nrm2[n - 1];

    // ---- bb: per interior i, closed form via the 6 scalars ----
    const float labA = labels[(size_t)b * N + 0];
    const float labT = labels[(size_t)b * N + (n - 1)];
    float bbp = 0.f;
    if (tid >= 1 && tid < n - 1) {
        float li = labels[(size_t)b * N + tid];
        float a  = (li - labA) / (labT - labA);
        float sig = a * (labT - li);
        float mI = (float)(n - 2);
        float dIS = (1.f - a) * P0 + a * Pe;
        float nI  = (1.f - a) * (1.f - a) * n00 + 2.f * a * (1.f - a) * N0E + a * a * nee;
        float val = S2 - 2.f * dIS + mI * nI;
        bbp = val / (2.f * sig);
    }
    const float bbv = blockReduceSum(bbp, s_red);

    // ---- logits s[i] = ip / (||f_i|| ||t||) / TEMP ----
    if (tid < N) s_sv[tid] = s_ip[tid] * rsqrtf(s_nrm2[tid] * tnorm2) * TEMP_INV;
    __syncthreads();

    // ---- loss: denom[a,b] = suffix sum of exp(L[a,.]) over |a-m| >= |a-b|
    //      (labels are equally spaced); row_max == 0 for valid rows. ----
    float tot = 0.f;
    const int a = tid;
    if (a < n) {
        const float sa = s_sv[a];
        const int maxd = max(a, n - 1 - a);
        float S = 0.f;
        for (int t = maxd; t >= 1; --t) {
            const int bl = a - t, br = a + t;
            float Ls = 0.f, cnt = 0.f;
            if (bl >= 0) { float Lv = -fabsf(sa - s_sv[bl]); S += __expf(Lv); Ls += Lv; cnt += 1.f; }
            if (br < n)  { float Lv = -fabsf(sa - s_sv[br]); S += __expf(Lv); Ls += Lv; cnt += 1.f; }
            tot += Ls - cnt * __logf(S);
        }
    }
    const float lossSum = blockReduceSum(tot, s_red);

    if (tid == 0) {
        const float nf = (float)n;
        ws[b]             = -lossSum / (nf * (nf - 1.f));
        ws[gridDim.x + b] = bbv;
    }
}

__global__ void rnc_finalize_kernel(const float* __restrict__ ws,
                                    float* __restrict__ out, int bs) {
    if (threadIdx.x == 0) {
        float l = 0.f, bb = 0.f;
        for (int i = 0; i < bs; ++i) { l += ws[i]; bb += ws[bs + i]; }
        out[0] = l / (float)bs;
        out[1] = bb / (float)bs;
    }
}

extern "C" void kernel_launch(void* const* d_in, const int* in_sizes, int n_in,
                              void* d_out, int out_size, void* d_ws, size_t ws_size,
                              hipStream_t stream) {
    const float* feat    = (const float*)d_in[0];  // (bs,N,D) f32
    const float* text    = (const float*)d_in[1];  // (bs,D)   f32
    const int*   nframes = (const int*)  d_in[2];  // (bs,)    i32
    const float* labels  = (const float*)d_in[3];  // (bs,N,1) f32

    const int bs = in_sizes[2];
    const int D  = in_sizes[1] / bs;
    const int N  = in_sizes[0] / (bs * D);

    float* ws = (float*)d_ws;

    rnc_per_sample_kernel<<<bs, NTHREADS, 0, stream>>>(feat, text, nframes, labels,
                                                       ws, N, D);
    rnc_finalize_kernel<<<1, 32, 0, stream>>>(ws, (float*)d_out, bs);
}